// GNNClassifier_58256936403610
// MI455X (gfx1250) — compile-verified
//
#include <hip/hip_runtime.h>
#include <math.h>

// ---------------------------------------------------------------------------
// GCN (2x GCNConv + mean-pool + MLP) for MI455X (gfx1250, wave32).
//
// Data-movement strategy (reasoned from MI455X specs, no runtime available):
//  * All per-node feature buffers (12.8 MB) are L2-resident (192 MB L2).
//  * Aggregation is PULL-based via a device-built CSR (col-sorted):
//    one wave per node, lane = feature, coalesced 128B row gathers that hit
//    L2, register accumulation, fused self-loop + bias + ReLU.  This avoids
//    ~205M global f32 atomics (which would be L2-atomic-unit bound, ~10x off
//    the bandwidth roofline).
//  * h1 @ W2 runs on the matrix pipe: V_WMMA_F32_16X16X4_F32, full fp32.
//  * Mean-pool exploits sorted `batch`: one wave per graph, binary-search
//    segment bounds, streaming reduction (atomic-free, deterministic).
// ---------------------------------------------------------------------------

#define HID 32
#define TPB 256
#define SCANB 1024
#define GRID(n) ((((n) + TPB - 1)) / TPB)

typedef float v2f __attribute__((ext_vector_type(2)));
typedef float v8f __attribute__((ext_vector_type(8)));

// ---------------- CSR construction ----------------

__global__ void zero_i32(int* __restrict__ p, int n) {
  int i = blockIdx.x * blockDim.x + threadIdx.x;
  if (i < n) p[i] = 0;
}

__global__ void count_edges(const int* __restrict__ col, int* __restrict__ cnt, int E) {
  int e = blockIdx.x * blockDim.x + threadIdx.x;
  if (e < E) atomicAdd(cnt + col[e], 1);
}

// Per-block exclusive scan (Hillis-Steele in LDS); block totals to blockSums.
__global__ void scan_block(const int* __restrict__ in, int* __restrict__ out,
                           int* __restrict__ blockSums, int n) {
  __shared__ int sh[TPB];
  int i = blockIdx.x * TPB + threadIdx.x;
  int v = (i < n) ? in[i] : 0;
  sh[threadIdx.x] = v;
  __syncthreads();
  for (int off = 1; off < TPB; off <<= 1) {
    int t = (threadIdx.x >= off) ? sh[threadIdx.x - off] : 0;
    __syncthreads();
    sh[threadIdx.x] += t;
    __syncthreads();
  }
  int incl = sh[threadIdx.x];
  if (i < n) out[i] = incl - v;                      // exclusive
  if (threadIdx.x == TPB - 1) blockSums[blockIdx.x] = incl;
}

// Exclusive scan of block totals (single block; serial fallback if huge).
__global__ void scan_sums(int* __restrict__ sums, int nb) {
  __shared__ int sh[SCANB];
  if (nb <= SCANB) {
    int v = ((int)threadIdx.x < nb) ? sums[threadIdx.x] : 0;
    sh[threadIdx.x] = v;
    __syncthreads();
    for (int off = 1; off < SCANB; off <<= 1) {
      int t = (threadIdx.x >= (unsigned)off) ? sh[threadIdx.x - off] : 0;
      __syncthreads();
      sh[threadIdx.x] += t;
      __syncthreads();
    }
    if ((int)threadIdx.x < nb) sums[threadIdx.x] = sh[threadIdx.x] - v;
  } else if (threadIdx.x == 0) {
    int run = 0;
    for (int i = 0; i < nb; ++i) { int v = sums[i]; sums[i] = run; run += v; }
  }
}

// rowptr[i] += blockSums[block]; also seed the fill cursors.
__global__ void scan_add(int* __restrict__ rowptr, const int* __restrict__ blockSums,
                         int* __restrict__ cursor, int n) {
  int i = blockIdx.x * TPB + threadIdx.x;
  if (i >= n) return;
  int v = rowptr[i] + blockSums[blockIdx.x];
  rowptr[i] = v;
  cursor[i] = v;
}

__global__ void set_total(int* __restrict__ rowptr, const int* __restrict__ cnt, int n) {
  if (blockIdx.x == 0 && threadIdx.x == 0) rowptr[n] = rowptr[n - 1] + cnt[n - 1];
}

__global__ void csr_fill(const int* __restrict__ row, const int* __restrict__ col,
                         int* __restrict__ cursor, int* __restrict__ csr_rows, int E) {
  int e = blockIdx.x * blockDim.x + threadIdx.x;
  if (e >= E) return;
  int slot = atomicAdd(cursor + col[e], 1);
  csr_rows[slot] = row[e];
}

// dinv = rsqrt(in_degree + 1 self-loop); always > 0.
__global__ void make_dinv(const int* __restrict__ cnt, float* __restrict__ dinv, int n) {
  int i = blockIdx.x * blockDim.x + threadIdx.x;
  if (i < n) dinv[i] = rsqrtf((float)cnt[i] + 1.0f);
}

// ---------------- GCN layers ----------------

// h[n][f] = x[n] * W1[f]  (fan_in == 1)
__global__ void layer1_h(const float* __restrict__ x, const float* __restrict__ W1,
                         float* __restrict__ h, int total) {
  int t = blockIdx.x * blockDim.x + threadIdx.x;
  if (t >= total) return;
  int node = t >> 5, f = t & (HID - 1);
  h[t] = x[node] * W1[f];
}

// Pull aggregation: one wave per node, lane = feature.
// out[c][f] = relu( sum_{r in N(c)} h[r][f]*dinv[r]*dinv[c] + h[c][f]*dinv[c]^2 + bias[f] )
__global__ void gather_agg(const int* __restrict__ rowptr, const int* __restrict__ csr_rows,
                           const float* __restrict__ dinv, const float* __restrict__ h,
                           const float* __restrict__ bias, float* __restrict__ out, int N) {
  int node = blockIdx.x * (TPB / 32) + (threadIdx.x >> 5);
  int lane = threadIdx.x & 31;
  if (node >= N) return;
  int beg = rowptr[node];
  int end = rowptr[node + 1];
  float dc = dinv[node];
  float acc = h[node * HID + lane] * dc * dc;        // self loop
  int j = beg;
  // 2-wide accumulation for a little ILP across the L2-latency gathers
  float acc2 = 0.0f;
  for (; j + 1 < end; j += 2) {
    int r0 = csr_rows[j];
    int r1 = csr_rows[j + 1];
    acc  += h[r0 * HID + lane] * (dinv[r0] * dc);
    acc2 += h[r1 * HID + lane] * (dinv[r1] * dc);
  }
  if (j < end) {
    int r = csr_rows[j];
    acc += h[r * HID + lane] * (dinv[r] * dc);
  }
  out[node * HID + lane] = fmaxf(acc + acc2 + bias[lane], 0.0f);
}

// hh[N,32] = h1[N,32] @ W2[32,32] via V_WMMA_F32_16X16X4_F32.
// One wave per 16-node tile; two 16x16 accumulators (cols 0..15 / 16..31);
// K=32 unrolled as 8 WMMAs of K=4.  (ISA 7.12.2 f32 fragment layouts.)
__global__ void wmma_matmul(const float* __restrict__ h1, const float* __restrict__ W2,
                            float* __restrict__ hh, int tiles) {
  int lane = threadIdx.x & 31;
  int tile = blockIdx.x * (TPB / 32) + (threadIdx.x >> 5);
  if (tile >= tiles) return;                 // wave-uniform: EXEC all-ones for WMMA
  int base = tile * 16;
  int m    = lane & 15;                      // row within tile (A) / col (B,D)
  int koff = (lane >> 4) * 2;                // 0 for lanes 0-15, 2 for lanes 16-31

  v8f c0 = {};
  v8f c1 = {};
  const float* arow = h1 + (size_t)(base + m) * HID + koff;

#pragma unroll
  for (int ks = 0; ks < 8; ++ks) {
    int k0 = ks * 4;
    v2f a = *reinterpret_cast<const v2f*>(arow + k0);
    const float* wb = W2 + (size_t)(k0 + koff) * HID + m;
    v2f b0, b1v;
    b0.x  = wb[0];        // K = k0+koff,   N = m
    b0.y  = wb[HID];      // K = k0+koff+1, N = m
    b1v.x = wb[16];       // cols 16..31
    b1v.y = wb[HID + 16];
    c0 = __builtin_amdgcn_wmma_f32_16x16x4_f32(false, a, false, b0,  (short)0, c0, false, false);
    c1 = __builtin_amdgcn_wmma_f32_16x16x4_f32(false, a, false, b1v, (short)0, c1, false, false);
  }

  int mhi = (lane >> 4) * 8;
#pragma unroll
  for (int r = 0; r < 8; ++r) {
    int node = base + mhi + r;
    hh[(size_t)node * HID + m]      = c0[r];
    hh[(size_t)node * HID + 16 + m] = c1[r];
  }
}

// Scalar fallback for ragged tail (N % 16 != 0); unused for N=100000.
__global__ void tail_matmul(const float* __restrict__ h1, const float* __restrict__ W2,
                            float* __restrict__ hh, int start, int N) {
  int t = blockIdx.x * blockDim.x + threadIdx.x;
  int total = (N - start) * HID;
  if (t >= total) return;
  int node = start + (t >> 5), f = t & (HID - 1);
  float acc = 0.0f;
  for (int k = 0; k < HID; ++k) acc += h1[node * HID + k] * W2[k * HID + f];
  hh[node * HID + f] = acc;
}

// ---------------- pooling + head ----------------

// batch is sorted: one wave per graph, binary-search [start,end), stream-sum.
__global__ void pool_mean(const float* __restrict__ h2, const int* __restrict__ batch,
                          float* __restrict__ pooled, int N, int G) {
  int g = blockIdx.x * (TPB / 32) + (threadIdx.x >> 5);
  int lane = threadIdx.x & 31;
  if (g >= G) return;
  int lo = 0, hi = N;
  while (lo < hi) { int mid = (lo + hi) >> 1; if (batch[mid] < g) lo = mid + 1; else hi = mid; }
  int start = lo;
  lo = start; hi = N;
  while (lo < hi) { int mid = (lo + hi) >> 1; if (batch[mid] <= g) lo = mid + 1; else hi = mid; }
  int end = lo;
  float s = 0.0f;
  for (int n = start; n < end; ++n) s += h2[n * HID + lane];
  float c = fmaxf((float)(end - start), 1.0f);
  pooled[g * HID + lane] = s / c;
}

__global__ void mlp_kernel(const float* __restrict__ pooled,
                           const float* __restrict__ Wl1, const float* __restrict__ bl1,
                           const float* __restrict__ Wl2, const float* __restrict__ bl2,
                           float* __restrict__ out, int G) {
  int g = blockIdx.x * blockDim.x + threadIdx.x;
  if (g >= G) return;
  float p[HID];
#pragma unroll
  for (int f = 0; f < HID; ++f) p[f] = pooled[g * HID + f];
  float acc = bl2[0];
#pragma unroll
  for (int j = 0; j < 16; ++j) {
    float z = bl1[j];
    for (int f = 0; f < HID; ++f) z += p[f] * Wl1[f * 16 + j];
    acc += fmaxf(z, 0.0f) * Wl2[j];
  }
  out[g] = 1.0f / (1.0f + expf(-acc));
}

// ---------------- launch ----------------

extern "C" void kernel_launch(void* const* d_in, const int* in_sizes, int n_in,
                              void* d_out, int out_size, void* d_ws, size_t ws_size,
                              hipStream_t stream) {
  const float* x    = (const float*)d_in[0];
  const int*   ei   = (const int*)  d_in[1];
  const int*   batc = (const int*)  d_in[2];
  const float* W1   = (const float*)d_in[4];
  const float* b1   = (const float*)d_in[5];
  const float* W2   = (const float*)d_in[6];
  const float* b2   = (const float*)d_in[7];
  const float* Wl1  = (const float*)d_in[8];
  const float* bl1  = (const float*)d_in[9];
  const float* Wl2  = (const float*)d_in[10];
  const float* bl2  = (const float*)d_in[11];

  const int N = in_sizes[0];       // x is [N,1]
  const int E = in_sizes[1] / 2;   // edge_index is [2,E]
  const int G = out_size;          // one sigmoid output per graph
  const int* row = ei;
  const int* col = ei + E;
  const int NT = N * HID;
  const int nBlocks = GRID(N);

  // workspace carve-out (256B aligned slabs)
  char* wsb = (char*)d_ws;
  size_t off = 0;
  auto alloc = [&](size_t bytes) { void* p = wsb + off; off += (bytes + 255) & ~(size_t)255; return p; };
  int*   cnt       = (int*)  alloc((size_t)N * 4);
  int*   rowptr    = (int*)  alloc((size_t)(N + 1) * 4);
  int*   blockSums = (int*)  alloc((size_t)nBlocks * 4);
  int*   cursor    = (int*)  alloc((size_t)N * 4);
  int*   csr_rows  = (int*)  alloc((size_t)E * 4);
  float* dinv      = (float*)alloc((size_t)N * 4);
  float* bufA      = (float*)alloc((size_t)NT * 4);  // h -> hh
  float* bufB      = (float*)alloc((size_t)NT * 4);  // h1 -> h2
  float* pooled    = (float*)alloc((size_t)G * HID * 4);
  (void)ws_size; (void)n_in;

  // ---- CSR build (one-time; only atomics in the whole pipeline) ----
  zero_i32<<<GRID(N), TPB, 0, stream>>>(cnt, N);
  count_edges<<<GRID(E), TPB, 0, stream>>>(col, cnt, E);
  scan_block<<<nBlocks, TPB, 0, stream>>>(cnt, rowptr, blockSums, N);
  scan_sums<<<1, SCANB, 0, stream>>>(blockSums, nBlocks);
  scan_add<<<nBlocks, TPB, 0, stream>>>(rowptr, blockSums, cursor, N);
  set_total<<<1, 1, 0, stream>>>(rowptr, cnt, N);
  make_dinv<<<GRID(N), TPB, 0, stream>>>(cnt, dinv, N);
  csr_fill<<<GRID(E), TPB, 0, stream>>>(row, col, cursor, csr_rows, E);

  // ---- layer 1: h = x*W1, pull-aggregate + bias + relu ----
  layer1_h<<<GRID(NT), TPB, 0, stream>>>(x, W1, bufA, NT);
  gather_agg<<<GRID(N * 32), TPB, 0, stream>>>(rowptr, csr_rows, dinv, bufA, b1, bufB, N);

  // ---- layer 2: hh = h1 @ W2 (WMMA f32), pull-aggregate + bias + relu ----
  int tiles = N / 16;
  if (tiles > 0)
    wmma_matmul<<<(tiles + (TPB / 32) - 1) / (TPB / 32), TPB, 0, stream>>>(bufB, W2, bufA, tiles);
  int tailStart = tiles * 16;
  if (tailStart < N)
    tail_matmul<<<GRID((N - tailStart) * HID), TPB, 0, stream>>>(bufB, W2, bufA, tailStart, N);
  gather_agg<<<GRID(N * 32), TPB, 0, stream>>>(rowptr, csr_rows, dinv, bufA, b2, bufB, N);

  // ---- mean pool (sorted batch, atomic-free) + MLP head ----
  pool_mean<<<(G + (TPB / 32) - 1) / (TPB / 32), TPB, 0, stream>>>(bufB, batc, pooled, N, G);
  mlp_kernel<<<GRID(G), TPB, 0, stream>>>(pooled, Wl1, bl1, Wl2, bl2, (float*)d_out, G);
}